// DenseCL_59734405153348
// MI455X (gfx1250) — compile-verified
//
#include <hip/hip_runtime.h>

typedef __attribute__((ext_vector_type(2))) float v2f;
typedef __attribute__((ext_vector_type(8))) float v8f;

#define TAU_INV 5.0f
#define DIM   128
#define SPOS  49
#define QN    65536
#define ROWS  65537   // 1 + Q
#define OUTW  50      // 1 + S

// ---------------------------------------------------------------------------
// Kernel 1: cosine (49x49, K=2048) -> argmax -> gather d_qm into padded ws
//           [bs][128][64] (zeros for z=49..63), plus pos row out[b,0,:].
// ---------------------------------------------------------------------------
__global__ __launch_bounds__(512)
void densecl_prep(const float* __restrict__ g_q, const float* __restrict__ g_k,
                  const float* __restrict__ d_q, const float* __restrict__ d_k,
                  const float* __restrict__ feat_q, const float* __restrict__ feat_k,
                  float* __restrict__ out, float* __restrict__ dqm_ws)
{
    __shared__ float kbuf[128 * SPOS];
    __shared__ float qbuf[128 * SPOS];
    __shared__ float cosb[SPOS * SPOS];
    __shared__ int   mi[SPOS];

    const int b   = blockIdx.x;
    const int tid = threadIdx.x;

    float acc[5] = {0.f, 0.f, 0.f, 0.f, 0.f};

    for (int c0 = 0; c0 < 2048; c0 += 128) {
        __syncthreads();
        const size_t base = ((size_t)b * 2048 + c0) * SPOS;
        for (int l = tid; l < 128 * SPOS; l += 512) {
            kbuf[l] = feat_k[base + l];
            qbuf[l] = feat_q[base + l];
        }
        __syncthreads();
        #pragma unroll
        for (int s = 0; s < 5; ++s) {
            int p = tid + s * 512;
            if (p < SPOS * SPOS) {
                int i = p % SPOS, j = p / SPOS;
                float a = acc[s];
                #pragma unroll 8
                for (int cc = 0; cc < 128; ++cc)
                    a += kbuf[cc * SPOS + i] * qbuf[cc * SPOS + j];
                acc[s] = a;
            }
        }
    }
    #pragma unroll
    for (int s = 0; s < 5; ++s) {
        int p = tid + s * 512;
        if (p < SPOS * SPOS) {
            int i = p % SPOS, j = p / SPOS;
            cosb[j * SPOS + i] = acc[s];
        }
    }
    __syncthreads();

    if (tid < SPOS) {
        float best = cosb[tid * SPOS];
        int   bi   = 0;
        for (int i = 1; i < SPOS; ++i) {
            float v = cosb[tid * SPOS + i];
            if (v > best) { best = v; bi = i; }   // first-occurrence argmax
        }
        mi[tid] = bi;
    }
    __syncthreads();

    // gather d_qm -> ws padded [128][64]  (regular stores: keep L2-resident)
    const size_t dq_base = (size_t)b * DIM * SPOS;
    float* ws = dqm_ws + (size_t)b * DIM * 64;
    for (int l = tid; l < DIM * 64; l += 512) {
        int d = l >> 6, j = l & 63;
        float v = 0.f;
        if (j < SPOS) v = d_q[dq_base + (size_t)d * SPOS + mi[j]];
        ws[l] = v;
    }

    // pos row: out[b,0,0] and out[b,0,1+z]
    float* orow = out + (size_t)b * ROWS * OUTW;
    if (tid < SPOS) {
        const int j = tid, m = mi[tid];
        float sum = 0.f;
        for (int d = 0; d < DIM; ++d)
            sum += d_k[dq_base + (size_t)d * SPOS + j] *
                   d_q[dq_base + (size_t)d * SPOS + m];
        __builtin_nontemporal_store(sum * TAU_INV, &orow[1 + j]);
    }
    if (tid == 56) {
        const float* gq = g_q + b * DIM;
        const float* gk = g_k + b * DIM;
        float sum = 0.f;
        for (int d = 0; d < DIM; ++d) sum += gq[d] * gk[d];
        __builtin_nontemporal_store(sum * TAU_INV, &orow[0]);
    }
}

// ---------------------------------------------------------------------------
// Kernel 2: per wave a 16-q tile. WMMA F32 16x16x4: A = queue_d^T (M=q,K=dim),
// B = d_qm from LDS (K=dim, N=z padded to 64). 32 K-steps x 4 N-tiles = 128
// v_wmma per wave. out_neg_g via VALU dot on lanes 0..15. Scale by 1/tau.
// Grid = (batch=32, qtile=512): all 32 batch-reuses of a queue slice launch
// back-to-back -> queue_d/queue_g stay L2-resident. Output stores are NT so
// the 419 MB write stream doesn't evict the queues from L2.
// ---------------------------------------------------------------------------
__global__ __launch_bounds__(256)
void densecl_main(const float* __restrict__ g_q,
                  const float* __restrict__ queue_g,
                  const float* __restrict__ queue_d,
                  const float* __restrict__ dqm_ws,
                  float* __restrict__ out)
{
    __shared__ float dqm[DIM * 64];   // [d][z] padded, 32 KB
    __shared__ float gql[DIM];

    const int b   = blockIdx.x;
    const int tid = threadIdx.x;

    // cooperative LDS fill (coalesced float4)
    {
        const float4* src = (const float4*)(dqm_ws + (size_t)b * DIM * 64);
        float4* dst = (float4*)dqm;
        for (int l = tid; l < DIM * 64 / 4; l += 256) dst[l] = src[l];
        if (tid < DIM) gql[tid] = g_q[b * DIM + tid];
    }
    __syncthreads();

    const int w    = tid >> 5;
    const int lane = tid & 31;
    const int qb   = blockIdx.y * 128 + w * 16;
    const int n    = lane & 15;
    const int q    = qb + n;
    const int koff = (lane >= 16) ? 2 : 0;
    const int hi   = (lane >= 16) ? 8 : 0;

    v8f acc0 = {}, acc1 = {}, acc2 = {}, acc3 = {};

    const float* aBase = queue_d + (size_t)koff * QN + q;
    #pragma unroll 4
    for (int ks = 0; ks < 32; ++ks) {
        const int k4 = ks * 4;
        const float* ap = aBase + (size_t)k4 * QN;
        v2f a; a.x = ap[0]; a.y = ap[QN];

        const float* bp = &dqm[(k4 + koff) * 64 + n];
        v2f b0; b0.x = bp[0];  b0.y = bp[64];
        v2f b1; b1.x = bp[16]; b1.y = bp[80];
        v2f b2; b2.x = bp[32]; b2.y = bp[96];
        v2f b3; b3.x = bp[48]; b3.y = bp[112];

        acc0 = __builtin_amdgcn_wmma_f32_16x16x4_f32(false, a, false, b0, (short)0, acc0, false, false);
        acc1 = __builtin_amdgcn_wmma_f32_16x16x4_f32(false, a, false, b1, (short)0, acc1, false, false);
        acc2 = __builtin_amdgcn_wmma_f32_16x16x4_f32(false, a, false, b2, (short)0, acc2, false, false);
        acc3 = __builtin_amdgcn_wmma_f32_16x16x4_f32(false, a, false, b3, (short)0, acc3, false, false);
    }

    // global-branch logits for this q-tile (lanes 0..15)
    float gacc = 0.f;
    if (lane < 16) {
        const float* qg = queue_g + q;
        #pragma unroll 8
        for (int d = 0; d < DIM; ++d) gacc += qg[(size_t)d * QN] * gql[d];
    }

    float* ob = out + (size_t)b * ROWS * OUTW;
    #pragma unroll
    for (int r = 0; r < 8; ++r) {
        const size_t row = (size_t)(1 + qb + hi + r) * OUTW;
        __builtin_nontemporal_store(acc0[r] * TAU_INV, &ob[row + 1 +  0 + n]);
        __builtin_nontemporal_store(acc1[r] * TAU_INV, &ob[row + 1 + 16 + n]);
        __builtin_nontemporal_store(acc2[r] * TAU_INV, &ob[row + 1 + 32 + n]);
        if (n == 0)
            __builtin_nontemporal_store(acc3[r] * TAU_INV, &ob[row + 1 + 48]);
    }
    if (lane < 16)
        __builtin_nontemporal_store(gacc * TAU_INV, &ob[(size_t)(1 + qb + lane) * OUTW]);
}

extern "C" void kernel_launch(void* const* d_in, const int* in_sizes, int n_in,
                              void* d_out, int out_size, void* d_ws, size_t ws_size,
                              hipStream_t stream) {
    const float* g_q     = (const float*)d_in[0];
    const float* g_k     = (const float*)d_in[1];
    const float* d_q     = (const float*)d_in[2];
    const float* d_k     = (const float*)d_in[3];
    const float* feat_q  = (const float*)d_in[4];
    const float* feat_k  = (const float*)d_in[5];
    const float* queue_g = (const float*)d_in[6];
    const float* queue_d = (const float*)d_in[7];
    float* out = (float*)d_out;
    float* ws  = (float*)d_ws;   // needs 32*128*64*4 = 1 MB

    densecl_prep<<<32, 512, 0, stream>>>(g_q, g_k, d_q, d_k, feat_q, feat_k, out, ws);
    densecl_main<<<dim3(32, 512), 256, 0, stream>>>(g_q, queue_g, queue_d, ws, out);
}